// RandomPermute_43671227466270
// MI455X (gfx1250) — compile-verified
//
#include <hip/hip_runtime.h>
#include <hip/hip_bf16.h>

#define DIM 512
#define ROWS_PER_BLOCK 4
#define BLOCK 256  // 8 waves of 32 on CDNA5

typedef float v4f __attribute__((ext_vector_type(4)));

// ---------------------------------------------------------------------------
// Kernel 1: extract permutation from the one-hot matrix Q.
// p[c] = r such that Q[r, c] == 1.  (D*D = 256K reads, negligible.)
// ---------------------------------------------------------------------------
__global__ void build_perm_kernel(const float* __restrict__ Q, int* __restrict__ p) {
    int c = blockIdx.x * blockDim.x + threadIdx.x;
    if (c >= DIM) return;
    int src = 0;
    for (int r = 0; r < DIM; ++r) {
        if (Q[(long)r * DIM + c] > 0.5f) src = r;
    }
    p[c] = src;
}

// ---------------------------------------------------------------------------
// Kernel 2: out[n, c] = X[n, p[c]]   (column gather, HBM-bandwidth bound)
//
// Per block: stage ROWS_PER_BLOCK rows of X into LDS with CDNA5 async
// global->LDS 128-bit loads (coalesced, tracked by ASYNCcnt), then gather
// from LDS and emit coalesced 128-bit non-temporal stores.
// ---------------------------------------------------------------------------
__global__ __launch_bounds__(BLOCK)
void permute_cols_kernel(const float* __restrict__ X,
                         const int* __restrict__ p,
                         float* __restrict__ out,
                         int n_rows) {
    __shared__ float tile[ROWS_PER_BLOCK * DIM];  // 8 KB
    __shared__ int   sidx[DIM];                   // 2 KB

    const int t = threadIdx.x;
    const long row0 = (long)blockIdx.x * ROWS_PER_BLOCK;
    if (row0 >= n_rows) return;
    const long base = row0 * DIM;

    // Stage the permutation indices (read once per block, L2-resident).
    for (int i = t; i < DIM; i += BLOCK) sidx[i] = p[i];

    // Async-stage ROWS_PER_BLOCK rows of X into LDS, 16 bytes per request.
    // ROWS_PER_BLOCK*DIM/4 = 512 float4s; 2 per thread.
    #pragma unroll
    for (int f = t; f < ROWS_PER_BLOCK * DIM / 4; f += BLOCK) {
        const float* gp = X + base + (long)f * 4;
        // AS(3) offset: low 32 bits of the generic (flat) shared address.
        unsigned lds_off = (unsigned)(size_t)(&tile[f * 4]);
        asm volatile("global_load_async_to_lds_b128 %0, %1, off"
                     :: "v"(lds_off), "v"(gp)
                     : "memory");
    }
    // Wait for this wave's async transfers; barrier covers the other waves'.
    asm volatile("s_wait_asynccnt 0" ::: "memory");
    __syncthreads();

    // Gather from LDS and write coalesced 16B, non-temporal (never re-read).
    #pragma unroll
    for (int f = t; f < ROWS_PER_BLOCK * DIM / 4; f += BLOCK) {
        const int row = f >> 7;          // f / (DIM/4)
        const int c0  = (f & 127) << 2;  // 4 * (f % (DIM/4))
        const float* lrow = &tile[row * DIM];
        v4f v;
        v.x = lrow[sidx[c0 + 0]];
        v.y = lrow[sidx[c0 + 1]];
        v.z = lrow[sidx[c0 + 2]];
        v.w = lrow[sidx[c0 + 3]];
        __builtin_nontemporal_store(v, (v4f*)(out + base + (long)f * 4));
    }
}

// ---------------------------------------------------------------------------
// Launcher
// ---------------------------------------------------------------------------
extern "C" void kernel_launch(void* const* d_in, const int* in_sizes, int n_in,
                              void* d_out, int out_size, void* d_ws, size_t ws_size,
                              hipStream_t stream) {
    const float* X = (const float*)d_in[0];
    const float* Q = (const float*)d_in[1];
    float* out = (float*)d_out;
    int* perm = (int*)d_ws;  // 512 ints = 2 KB of workspace

    const int n_rows = in_sizes[0] / DIM;  // N = 262144

    build_perm_kernel<<<(DIM + BLOCK - 1) / BLOCK, BLOCK, 0, stream>>>(Q, perm);

    const int n_blocks = (n_rows + ROWS_PER_BLOCK - 1) / ROWS_PER_BLOCK;
    permute_cols_kernel<<<n_blocks, BLOCK, 0, stream>>>(X, perm, out, n_rows);
}